// BaseSegHead_48292612276572
// MI455X (gfx1250) — compile-verified
//
#include <hip/hip_runtime.h>

typedef __attribute__((ext_vector_type(2))) float v2f;
typedef __attribute__((ext_vector_type(8))) float v8f;

#define BATCH   4
#define N_PER   300
#define IN_DIM  256
#define KD      32
#define HW      25600          // 160*160
#define N_TILE  128            // spatial columns per workgroup
#define KW_STRIDE 260          // 256 + 4 pad (260 % 64 == 4 -> conflict-free)
#define Q_ROWS  304            // 300 padded to 19*16
#define Q_STRIDE 33            // 32 + 1 pad
#define KT_STRIDE 132          // 128 + 4 pad

__device__ __forceinline__ v8f wmma_f32_4(v2f a, v2f b, v8f c) {
    // D = A(16x4,f32) * B(4x16,f32) + C(16x16,f32)
    return __builtin_amdgcn_wmma_f32_16x16x4_f32(
        /*neg_a=*/false, a, /*neg_b=*/false, b,
        /*c_mod=*/(short)0, c, /*reuse_a=*/false, /*reuse_b=*/false);
}

// ---------------------------------------------------------------------------
// Kernel 1: qry projection  Q[1200,32] = in_feats @ qry_w^T + qry_b
// 20 MFLOP total -> simple scalar kernel, negligible runtime.
// ---------------------------------------------------------------------------
__global__ __launch_bounds__(256) void qry_proj_kernel(
    const float* __restrict__ in_feats,   // [1200,256]
    const float* __restrict__ qry_w,      // [32,256]
    const float* __restrict__ qry_b,      // [32]
    float* __restrict__ Q)                // [1200,32]
{
    int idx = blockIdx.x * 256 + threadIdx.x;      // 1200*32 = 38400 threads
    int row = idx >> 5;
    int col = idx & 31;
    const float* f = in_feats + row * IN_DIM;
    const float* w = qry_w   + col * IN_DIM;
    float acc = qry_b[col];
#pragma unroll 8
    for (int k = 0; k < IN_DIM; ++k) acc = fmaf(f[k], w[k], acc);
    Q[idx] = acc;
}

// ---------------------------------------------------------------------------
// Kernel 2: fused  key_map = key_w @ feat_map[b] + key_b   (stage 1, in LDS)
//                  out     = Q[b] @ key_map                (stage 2)
// One workgroup = (batch b, 128-wide spatial tile). 8 waves of 32.
// ---------------------------------------------------------------------------
__global__ __launch_bounds__(256) void fused_seg_kernel(
    const float* __restrict__ FM,   // [4,256,160,160]
    const float* __restrict__ KW,   // [32,256]
    const float* __restrict__ KB,   // [32]
    const float* __restrict__ Q,    // [1200,32]
    float* __restrict__ OUT)        // [1200,160,160]
{
    __shared__ float s_kw[KD * KW_STRIDE];      // key_w cache   (33.3 KB)
    __shared__ float s_q [Q_ROWS * Q_STRIDE];   // padded Q[b]   (40.1 KB)
    __shared__ float s_kt[KD * KT_STRIDE];      // key tile      (16.9 KB)

    const int b   = blockIdx.y;
    const int n0  = blockIdx.x * N_TILE;
    const int tid = threadIdx.x;

    // ---- preload key_w into LDS (padded stride) ----
    for (int i = tid; i < KD * IN_DIM; i += 256) {
        int r = i >> 8, cidx = i & 255;
        s_kw[r * KW_STRIDE + cidx] = KW[i];
    }
    // ---- preload Q[b] into LDS, zero-pad rows 300..303 ----
    const float* Qb = Q + (size_t)b * N_PER * KD;
    for (int i = tid; i < Q_ROWS * KD; i += 256) {
        int r = i >> 5, cidx = i & 31;
        s_q[r * Q_STRIDE + cidx] = (r < N_PER) ? Qb[i] : 0.0f;
    }
    __syncthreads();

    const int wv   = tid >> 5;       // wave id 0..7  == column tile id
    const int lane = tid & 31;
    const int lh   = lane >> 4;      // half-wave selector
    const int l16  = lane & 15;
    const int ct   = wv;             // this wave's 16-column strip
    const int ncol = ct * 16 + l16;  // column within the 128-wide tile

    // ================= Stage 1: key tile = key_w @ FM(:,tile) + bias =======
    {
        const float* FMb = FM + (size_t)b * IN_DIM * HW + (size_t)(n0 + ncol);
        v8f c0 = {};    // rows 0..15 of key_map
        v8f c1 = {};    // rows 16..31
#pragma unroll 4
        for (int k0 = 0; k0 < IN_DIM; k0 += 4) {
            const int ka = k0 + 2 * lh;
            v2f a0, a1, bb;
            a0[0] = s_kw[l16 * KW_STRIDE + ka];
            a0[1] = s_kw[l16 * KW_STRIDE + ka + 1];
            a1[0] = s_kw[(16 + l16) * KW_STRIDE + ka];
            a1[1] = s_kw[(16 + l16) * KW_STRIDE + ka + 1];
            const float* p = FMb + (size_t)ka * HW;   // coalesced 64B/half-wave
            bb[0] = p[0];
            bb[1] = p[HW];
            c0 = wmma_f32_4(a0, bb, c0);
            c1 = wmma_f32_4(a1, bb, c1);
        }
#pragma unroll
        for (int j = 0; j < 8; ++j) {
            const int r = j + 8 * lh;
            s_kt[r        * KT_STRIDE + ncol] = c0[j] + KB[r];
            s_kt[(16 + r) * KT_STRIDE + ncol] = c1[j] + KB[16 + r];
        }
    }
    __syncthreads();

    // ================= Stage 2: out tile = Q[b] @ key tile ================
    const float* ktc = s_kt + ncol;                       // column base in LDS
    const size_t obase = (size_t)b * N_PER * HW + (size_t)(n0 + ncol);
#pragma unroll 1
    for (int mt = 0; mt < Q_ROWS / 16; ++mt) {            // 19 row tiles
        v8f c = {};
#pragma unroll
        for (int k0 = 0; k0 < KD; k0 += 4) {
            const int ka = k0 + 2 * lh;
            v2f a, bb;
            a[0] = s_q[(mt * 16 + l16) * Q_STRIDE + ka];
            a[1] = s_q[(mt * 16 + l16) * Q_STRIDE + ka + 1];
            bb[0] = ktc[ka * KT_STRIDE];
            bb[1] = ktc[(ka + 1) * KT_STRIDE];
            c = wmma_f32_4(a, bb, c);
        }
#pragma unroll
        for (int j = 0; j < 8; ++j) {
            const int m = mt * 16 + j + 8 * lh;
            if (m < N_PER)
                OUT[obase + (size_t)m * HW] = c[j];
        }
    }
}

// ---------------------------------------------------------------------------
extern "C" void kernel_launch(void* const* d_in, const int* in_sizes, int n_in,
                              void* d_out, int out_size, void* d_ws, size_t ws_size,
                              hipStream_t stream) {
    const float* in_feats = (const float*)d_in[0];
    const float* feat_map = (const float*)d_in[1];
    const float* qry_w    = (const float*)d_in[2];
    const float* qry_b    = (const float*)d_in[3];
    const float* key_w    = (const float*)d_in[4];
    const float* key_b    = (const float*)d_in[5];
    float* out = (float*)d_out;
    float* Q   = (float*)d_ws;   // 1200*32 floats = 153.6 KB scratch

    qry_proj_kernel<<<(BATCH * N_PER * KD) / 256, 256, 0, stream>>>(
        in_feats, qry_w, qry_b, Q);

    dim3 grid(HW / N_TILE, BATCH);   // (200, 4)
    fused_seg_kernel<<<grid, 256, 0, stream>>>(feat_map, key_w, key_b, Q, out);
}